// VeinStemLoss_52175262712344
// MI455X (gfx1250) — compile-verified
//
#include <hip/hip_runtime.h>
#include <math.h>

#define NB      32           // B
#define NC      16           // C
#define NK      256          // K
#define NHW     (256 * 256)  // H*W
#define NROWS   (NB * NK)    // 8192
#define TPB     64           // 2 waves/block -> spread waves across many WGPs/SIMDs
#define NBLK    (NROWS / TPB) // 128
#define PI_F    3.14159265358979323846f

// ---- CDNA5 async global->LDS copy via inline asm (ISA: GLOBAL_LOAD_ASYNC_TO_LDS_B128,
// tracked with ASYNCcnt). vdst_lds holds the wave-relative LDS byte address
// (= low 32 bits of a generic pointer into __shared__); vaddr is the 64-bit
// global address.
__device__ __forceinline__ void async_copy_b128_to_lds(const float* gsrc, float* lds_dst) {
#if defined(__HIP_DEVICE_COMPILE__)
    unsigned lds_off = (unsigned)(unsigned long long)(const void*)lds_dst;
    asm volatile("global_load_async_to_lds_b128 %0, %1, off"
                 :
                 : "v"(lds_off), "v"(gsrc)
                 : "memory");
#else
    (void)gsrc; (void)lds_dst;
#endif
}

__device__ __forceinline__ void wait_asynccnt0() {
#if defined(__HIP_DEVICE_COMPILE__)
#if defined(__has_builtin) && __has_builtin(__builtin_amdgcn_s_wait_asynccnt)
    __builtin_amdgcn_s_wait_asynccnt(0);
#else
    asm volatile("s_wait_asynccnt 0" ::: "memory");
#endif
#endif
}

// One thread per (b,k) row: gather 16 channels of pred (strided by H*W),
// pull the contiguous 16-float target row via async copy to LDS, do the
// polar<->cartesian projection math, and reduce |pred - proj| within the
// block. Deterministic per-block partials. (The compiler sinks the pred
// gathers under the mask test, skipping ~50% of the scattered traffic.)
__global__ __launch_bounds__(TPB) void vein_rows_kernel(
    const float* __restrict__ output,   // (B, C, H, W)
    const int*   __restrict__ mask,     // (B, K)
    const int*   __restrict__ ind,      // (B, K)
    const float* __restrict__ target,   // (B, K, C)
    float*       __restrict__ partials) // (NBLK, 2)
{
    __shared__ float lds_t[TPB * NC];   // 4 KB: this block's target rows
    __shared__ float red_l[TPB / 32], red_c[TPB / 32];

    const int tid = threadIdx.x;
    const int row = blockIdx.x * TPB + tid;   // 0 .. NROWS-1
    const int b   = row >> 8;                 // K == 256

    const float* trow = target + (size_t)row * NC;
    float* myl = &lds_t[tid * NC];

    // Kick off async copy of this thread's 64-byte target row into LDS.
    async_copy_b128_to_lds(trow + 0,  myl + 0);
    async_copy_b128_to_lds(trow + 4,  myl + 4);
    async_copy_b128_to_lds(trow + 8,  myl + 8);
    async_copy_b128_to_lds(trow + 12, myl + 12);

    const int m   = mask[row];
    const int idx = ind[row];

    // Scattered gather of pred channels (stride H*W floats).
    float pr[NC];
    const float* obase = output + (size_t)b * NC * NHW + (size_t)idx;
#pragma unroll
    for (int c = 0; c < NC; ++c) pr[c] = obase[(size_t)c * NHW];

    // Wait for the async target copy, then read this thread's own LDS region.
    wait_asynccnt0();
    float tg[NC];
#pragma unroll
    for (int c = 0; c < NC; ++c) tg[c] = myl[c];

    float loss = 0.0f;
    float cnt  = 0.0f;
    if (m != 0) {
        cnt = (float)NC;  // maskf is broadcast over all 16 channels

        // polar (r, deg) -> cartesian for pred & target (8 keypoints)
        float px[8], py[8], tx[8], ty[8];
#pragma unroll
        for (int j = 0; j < 8; ++j) {
            float pa = pr[2 * j + 1] * (PI_F / 180.0f);
            px[j] = pr[2 * j] * cosf(pa);
            py[j] = pr[2 * j] * sinf(pa);
            float ta = tg[2 * j + 1] * (PI_F / 180.0f);
            tx[j] = tg[2 * j] * cosf(ta);
            ty[j] = tg[2 * j] * sinf(ta);
        }

        float qx[8], qy[8];
        // TRUE_KP = {0,3,7}: copy target
        qx[0] = tx[0]; qy[0] = ty[0];
        qx[3] = tx[3]; qy[3] = ty[3];
        qx[7] = tx[7]; qy[7] = ty[7];

        // SEG = {1,2,4,5,6}: project pred point onto adjacent target segments
        const int segs[5] = {1, 2, 4, 5, 6};
#pragma unroll
        for (int s = 0; s < 5; ++s) {
            const int j = segs[s];
            const float ax = tx[j - 1], ay = ty[j - 1];
            const float mx = tx[j],     my = ty[j];
            const float bx = tx[j + 1], by = ty[j + 1];
            const float pxx = px[j],    pyy = py[j];

            // segment a->m
            float abx = mx - ax, aby = my - ay;
            float apx = pxx - ax, apy = pyy - ay;
            float t1 = (apx * abx + apy * aby) / (abx * abx + aby * aby);
            t1 = t1 < 0.0f ? 0.0f : (t1 > 1.0f ? 1.0f : t1);
            float c1x = ax + t1 * abx, c1y = ay + t1 * aby;
            float e1x = pxx - c1x, e1y = pyy - c1y;
            float d1 = sqrtf(e1x * e1x + e1y * e1y);

            // segment m->b
            float b2x = bx - mx, b2y = by - my;
            float p2x = pxx - mx, p2y = pyy - my;
            float t2 = (p2x * b2x + p2y * b2y) / (b2x * b2x + b2y * b2y);
            t2 = t2 < 0.0f ? 0.0f : (t2 > 1.0f ? 1.0f : t2);
            float c2x = mx + t2 * b2x, c2y = my + t2 * b2y;
            float e2x = pxx - c2x, e2y = pyy - c2y;
            float d2 = sqrtf(e2x * e2x + e2y * e2y);

            const bool take2 = d2 < d1;
            float sx = take2 ? c2x : c1x;
            float sy = take2 ? c2y : c1y;
            const bool degen = (ax == bx) && (ay == by);
            qx[j] = degen ? mx : sx;
            qy[j] = degen ? my : sy;
        }

        // cartesian -> polar, accumulate L1 against (masked) pred
#pragma unroll
        for (int j = 0; j < 8; ++j) {
            float dist = sqrtf(qx[j] * qx[j] + qy[j] * qy[j]);
            float ang  = atan2f(qy[j], qx[j]) * (180.0f / PI_F);
            if (ang < 0.0f) ang += 360.0f;
            loss += fabsf(pr[2 * j] - dist) + fabsf(pr[2 * j + 1] - ang);
        }
    }

    // wave32 reduction, then cross-wave via LDS (deterministic)
    const int lane = tid & 31;
    const int wid  = tid >> 5;   // 0..1
#pragma unroll
    for (int off = 16; off > 0; off >>= 1) {
        loss += __shfl_down(loss, off, 32);
        cnt  += __shfl_down(cnt,  off, 32);
    }
    if (lane == 0) { red_l[wid] = loss; red_c[wid] = cnt; }
    __syncthreads();
    if (tid == 0) {
        float L = 0.0f, Cc = 0.0f;
#pragma unroll
        for (int w = 0; w < TPB / 32; ++w) { L += red_l[w]; Cc += red_c[w]; }
        partials[blockIdx.x * 2 + 0] = L;
        partials[blockIdx.x * 2 + 1] = Cc;
    }
}

// Single-wave finalize over the NBLK block partials (4 per lane, then shfl).
__global__ __launch_bounds__(32) void vein_finalize_kernel(
    const float* __restrict__ partials, float* __restrict__ out)
{
    const int tid = threadIdx.x; // 0..31
    float L = 0.0f, Cc = 0.0f;
#pragma unroll
    for (int g = 0; g < NBLK / 32; ++g) {
        const int blk = g * 32 + tid;
        L  += partials[blk * 2 + 0];
        Cc += partials[blk * 2 + 1];
    }
#pragma unroll
    for (int off = 16; off > 0; off >>= 1) {
        L  += __shfl_down(L,  off, 32);
        Cc += __shfl_down(Cc, off, 32);
    }
    if (tid == 0) out[0] = L / (Cc + 1e-4f);
}

extern "C" void kernel_launch(void* const* d_in, const int* in_sizes, int n_in,
                              void* d_out, int out_size, void* d_ws, size_t ws_size,
                              hipStream_t stream) {
    const float* output = (const float*)d_in[0]; // (32,16,256,256) f32
    const int*   mask   = (const int*)  d_in[1]; // (32,256) i32
    const int*   ind    = (const int*)  d_in[2]; // (32,256) i32
    const float* target = (const float*)d_in[3]; // (32,256,16) f32
    float*       out    = (float*)d_out;         // scalar loss
    float*       partials = (float*)d_ws;        // NBLK * 2 floats

    vein_rows_kernel<<<NBLK, TPB, 0, stream>>>(output, mask, ind, target, partials);
    vein_finalize_kernel<<<1, 32, 0, stream>>>(partials, out);
}